// FeedForwardNet_1726576856803
// MI455X (gfx1250) — compile-verified
//
#include <hip/hip_runtime.h>
#include <math.h>

// ---------------------------------------------------------------------------
// NEAT-style layered gather-MLP for MI455X (gfx1250).
// Bandwidth-bound (134 MB streamed / 23.3 TB/s ~ 5.8us); batched dot products
// implemented as diag(W_tile x G_tile^T) with V_WMMA_F32_16X16X4_F32.
// ---------------------------------------------------------------------------

typedef __attribute__((ext_vector_type(2))) float v2f;
typedef __attribute__((ext_vector_type(8))) float v8f;

#define N_IN        4096
#define N_LAYERS    8
#define PER_LAYER   16384
#define FAN_IN      128
#define SCALE_ACT   4.9f

#define TILE_M          16          // neurons per wave-tile (WMMA M)
#define ROW_PITCH       132         // LDS row pitch in floats: 132%64=4 -> conflict-free
#define WAVES_PER_BLOCK 2
#define BLOCK_THREADS   (32 * WAVES_PER_BLOCK)
#define TILES_PER_LAYER (PER_LAYER / TILE_M)            // 1024
#define BLOCKS_PER_LAYER (TILES_PER_LAYER / WAVES_PER_BLOCK) // 512

// buf layout: [0]=bias(1.0), [1..N_IN]=inputs, then 8 layers of 16384 outputs.

__global__ void neat_init_buf(const float* __restrict__ x, float* __restrict__ buf) {
    int i = blockIdx.x * blockDim.x + threadIdx.x;
    if (i == 0) buf[0] = 1.0f;
    if (i < N_IN) buf[1 + i] = x[i];
}

__global__ __launch_bounds__(BLOCK_THREADS)
void neat_layer_kernel(const float* __restrict__ W,    // [PER_LAYER][FAN_IN] this layer
                       const int*   __restrict__ IDX,  // [PER_LAYER][FAN_IN] this layer
                       float*       buf,               // node buffer (read < out_base, write tile)
                       int          out_base,          // 1 + N_IN + l*PER_LAYER
                       float* __restrict__ out_final)  // non-null only on last layer
{
    // Per-wave private LDS: W-tile + G-tile, each 16 rows x 132-float pitch.
    __shared__ __align__(16) float lds[WAVES_PER_BLOCK * 2 * TILE_M * ROW_PITCH];

    const int lane = threadIdx.x & 31;
    const int wave = threadIdx.x >> 5;
    float* wtile = &lds[wave * 2 * TILE_M * ROW_PITCH];
    float* gtile = wtile + TILE_M * ROW_PITCH;

    const int tile = blockIdx.x * WAVES_PER_BLOCK + wave;  // 0..1023
    const int n0   = tile * TILE_M;                        // first neuron of tile

    // ---- Stage: coalesced weight/index loads + L2 gathers -> LDS tiles ----
    const int c4 = lane * 4;  // column base 0..124, 4 floats per lane per row
    #pragma unroll 4
    for (int r = 0; r < TILE_M; ++r) {
        const size_t rowoff = (size_t)(n0 + r) * FAN_IN + c4;
        const float4 w4 = *(const float4*)(W + rowoff);
        const int4   i4 = *(const int4*)(IDX + rowoff);
        float4 g4;
        g4.x = buf[i4.x];
        g4.y = buf[i4.y];
        g4.z = buf[i4.z];
        g4.w = buf[i4.w];
        *(float4*)(wtile + r * ROW_PITCH + c4) = w4;
        *(float4*)(gtile + r * ROW_PITCH + c4) = g4;
    }
    __syncthreads();  // waits dscnt/loadcnt; tiles are wave-private but this is the safe sync

    // ---- Compute: 32 x V_WMMA_F32_16X16X4_F32, accumulate full 16x16 D ----
    // A (16x4 MxK) lane layout: lanes 0-15 hold {A[m][k0],A[m][k1]}, lanes 16-31 {A[m][k2],A[m][k3]}
    // B (4x16 KxN) lane layout: lanes 0-15 hold {B[k0][n],B[k1][n]}, lanes 16-31 {B[k2][n],B[k3][n]}
    // With B = G^T both operands read LDS at: row=(lane&15), kofs=2*(lane>>4).
    const int   mrow = lane & 15;
    const int   kofs = (lane >> 4) << 1;
    const float* wr = wtile + mrow * ROW_PITCH + kofs;
    const float* gr = gtile + mrow * ROW_PITCH + kofs;

    v8f acc = {0.f, 0.f, 0.f, 0.f, 0.f, 0.f, 0.f, 0.f};
    #pragma unroll
    for (int c = 0; c < FAN_IN / 4; ++c) {
        v2f a = *(const v2f*)(wr + 4 * c);
        v2f b = *(const v2f*)(gr + 4 * c);
        // (neg_a, A, neg_b, B, c_mod, C, reuse_a, reuse_b)
        acc = __builtin_amdgcn_wmma_f32_16x16x4_f32(
            false, a, false, b, (short)0, acc, false, false);
    }

    // ---- Extract diag(D) through LDS (reuse wtile; same-wave LDS is in-order) ----
    // C/D layout: VGPR v -> rows M=v (lanes 0-15, N=lane) and M=v+8 (lanes 16-31, N=lane-16)
    #pragma unroll
    for (int v = 0; v < 8; ++v) {
        const int m = (lane < 16) ? v : (v + 8);
        wtile[m * TILE_M + (lane & 15)] = acc[v];
    }

    if (lane < TILE_M) {
        const float pre = SCALE_ACT * wtile[lane * TILE_M + lane];
        const float s   = 1.0f / (1.0f + expf(-pre));
        buf[out_base + n0 + lane] = s;
        if (out_final) out_final[n0 + lane] = s;
    }
}

extern "C" void kernel_launch(void* const* d_in, const int* in_sizes, int n_in,
                              void* d_out, int out_size, void* d_ws, size_t ws_size,
                              hipStream_t stream) {
    const float* x       = (const float*)d_in[0];  // [1, N_IN]
    const float* weights = (const float*)d_in[1];  // [N_LAYERS, PER_LAYER, FAN_IN]
    const int*   src_idx = (const int*)d_in[2];    // [N_LAYERS, PER_LAYER, FAN_IN]
    float*       out     = (float*)d_out;          // [1, PER_LAYER]
    float*       buf     = (float*)d_ws;           // node buffer, ~528 KB

    (void)in_sizes; (void)n_in; (void)out_size; (void)ws_size;

    neat_init_buf<<<(N_IN + 255) / 256, 256, 0, stream>>>(x, buf);

    for (int l = 0; l < N_LAYERS; ++l) {
        const size_t loff = (size_t)l * PER_LAYER * FAN_IN;
        neat_layer_kernel<<<BLOCKS_PER_LAYER, BLOCK_THREADS, 0, stream>>>(
            weights + loff,
            src_idx + loff,
            buf,
            1 + N_IN + l * PER_LAYER,
            (l == N_LAYERS - 1) ? out : nullptr);
    }
}